// CapsuleLayer_9663676416500
// MI455X (gfx1250) — compile-verified
//
#include <hip/hip_runtime.h>
#include <math.h>

// CapsNet dynamic routing, fused for MI455X (gfx1250, wave32).
//
// u_hat[b,n,i,d] = sum_k W[n,i,d,k] * x[b,i,k]   (never materialized: 151 MB)
// Routing logits are linear in accumulated v:  b_t[n,i] = <u_hat[n,i,:], Vacc_t[n,:]>
// so we only keep Vacc[B,N,D] + s[B,N,D] (256 KB workspace) and recompute u_hat
// with V_WMMA_F32_16X16X4_F32 (batch tile of 16 = WMMA M dimension).
// u_hat tiles stay live in VGPRs across the softmax barrier; all W/LDS addresses
// are base-pointer + immediate-offset so the n-loop fully unrolls.

typedef __attribute__((ext_vector_type(2))) float v2f;
typedef __attribute__((ext_vector_type(8))) float v8f;

#define EPSQ 1e-7f

#define Bsz  64
#define Ncap 32
#define Icap 1152
#define Dcap 16
#define Kdin 8
#define BT   16            // batch tile = WMMA M
#define IC   16            // input capsules per chunk
#define NCHUNK (Icap / IC) // 72
#define NPW  4             // output capsules per wave (8 waves * 4 = 32)
#define WSTR_N (Icap * Dcap * Kdin)  // 147456 floats between consecutive n
#define WSTR_I (Dcap * Kdin)         // 128 floats between consecutive i

__global__ void caps_zero(float* __restrict__ p, int n) {
    int t = blockIdx.x * blockDim.x + threadIdx.x;
    if (t < n) p[t] = 0.0f;
}

__launch_bounds__(256)
__global__ void caps_pass(const float* __restrict__ x,
                          const float* __restrict__ W,
                          const float* __restrict__ Vacc,
                          float* __restrict__ sbuf)
{
    __shared__ __attribute__((aligned(16))) float xs[IC][BT][Kdin];   // 8 KB
    __shared__ __attribute__((aligned(16))) float vs[Ncap][Dcap][BT]; // 32 KB [n][d][b]
    __shared__ __attribute__((aligned(16))) float ls[BT][Ncap];       // 2 KB

    const int t    = threadIdx.x;
    const int wave = t >> 5;
    const int L    = t & 31;
    const int g    = L >> 4;
    const int ln   = L & 15;
    const int B0   = blockIdx.y * BT;
    const int i0   = blockIdx.x * IC;
    const int kbase = 2 * g;

    // ---- stage Vacc tile as vs[n][d][b]: coalesced global, immediate LDS offsets ----
    {
        float*       lb = &vs[0][0][0] + ((t >> 4) * (Dcap * BT) + (t & 15) * BT);
        const float* gb = Vacc + (size_t)B0 * (Ncap * Dcap) + t;
        #pragma unroll
        for (int h = 0; h < 2; ++h)
            #pragma unroll
            for (int jb = 0; jb < BT; ++jb)
                lb[h * (16 * Dcap * BT) + jb] = gb[h * 256 + jb * (Ncap * Dcap)];
    }
    // ---- stage x tile xs[il][m][k] ----
    {
        float*       lb = &xs[0][0][0] + ((t >> 7) * 128 + (t & 127));
        const float* gb = x + (size_t)(B0 + ((t >> 3) & 15)) * (Icap * Kdin)
                            + (size_t)(i0 + (t >> 7)) * Kdin + (t & 7);
        #pragma unroll
        for (int j = 0; j < 8; ++j)
            lb[j * 256] = gb[j * 16];     // il advances by 2 per j
    }
    __syncthreads();

    float sAcc[NPW][8] = {};

    // Per-lane base pointers; all inner offsets are compile-time immediates.
    const float* wcur = W + ((size_t)(wave * NPW * Icap + i0) * Dcap + ln) * Kdin + kbase;
    const float* vl   = &vs[wave * NPW][ln][8 * g];

    for (int il = 0; il < IC; ++il) {
        v2f a0 = *(const v2f*)&xs[il][ln][kbase];
        v2f a1 = *(const v2f*)&xs[il][ln][4 + kbase];

        float uh[NPW][8];   // u_hat[b = r + 8g][d = ln], live across phases
        float tot[NPW];

        // ---- phase 1: u_hat via WMMA; logit = <u_hat, Vacc> via merge-tree ----
        #pragma unroll
        for (int np = 0; np < NPW; ++np) {
            const float* wp = wcur + (size_t)np * WSTR_N;
            __builtin_prefetch(wp + WSTR_I, 0, 3);        // next i, near cache
            v2f b0 = *(const v2f*)(wp);                    // k = kbase, kbase+1
            v2f b1 = *(const v2f*)(wp + 4);                // k = kbase+4, kbase+5
            v8f c = {};
            c = __builtin_amdgcn_wmma_f32_16x16x4_f32(false, a0, false, b0, (short)0, c, false, false);
            c = __builtin_amdgcn_wmma_f32_16x16x4_f32(false, a1, false, b1, (short)0, c, false, false);

            float lg[8];
            #pragma unroll
            for (int r = 0; r < 8; ++r) {
                uh[np][r] = c[r];
                lg[r] = c[r] * vl[np * (Dcap * BT) + r];   // 8 consecutive floats
            }
            // merge-tree: 8 regs x 16 lanes -> sum for reg r lands on lane r
            float m1[4];
            #pragma unroll
            for (int r = 0; r < 4; ++r) {
                float sel = (ln & 1) ? lg[2 * r + 1] : lg[2 * r];
                float oth = (ln & 1) ? lg[2 * r]     : lg[2 * r + 1];
                m1[r] = sel + __shfl_xor(oth, 1, 32);
            }
            float m2[2];
            #pragma unroll
            for (int r = 0; r < 2; ++r) {
                float sel = (ln & 2) ? m1[2 * r + 1] : m1[2 * r];
                float oth = (ln & 2) ? m1[2 * r]     : m1[2 * r + 1];
                m2[r] = sel + __shfl_xor(oth, 2, 32);
            }
            float sel3 = (ln & 4) ? m2[1] : m2[0];
            float oth3 = (ln & 4) ? m2[0] : m2[1];
            float m3 = sel3 + __shfl_xor(oth3, 4, 32);
            tot[np] = m3 + __shfl_xor(m3, 8, 32);          // lane l: sum for r = l&7
        }
        // single predicated vector store of 4 logits (n = wave*4 .. +3)
        if (ln < 8) {
            float4 st; st.x = tot[0]; st.y = tot[1]; st.z = tot[2]; st.w = tot[3];
            *(float4*)&ls[ln + 8 * g][wave * NPW] = st;
        }
        __syncthreads();

        // ---- phase 2: softmax over n (16 lane-groups <-> 16 b rows) ----
        {
            int bsm = 2 * wave + g;
            float2 l2 = *(const float2*)&ls[bsm][2 * ln];
            float mx = fmaxf(l2.x, l2.y);
            #pragma unroll
            for (int msk = 1; msk < 16; msk <<= 1) mx = fmaxf(mx, __shfl_xor(mx, msk, 32));
            float e0 = __expf(l2.x - mx), e1 = __expf(l2.y - mx);
            float sm = e0 + e1;
            #pragma unroll
            for (int msk = 1; msk < 16; msk <<= 1) sm += __shfl_xor(sm, msk, 32);
            float inv = 1.0f / sm;
            float2 c2; c2.x = e0 * inv; c2.y = e1 * inv;
            *(float2*)&ls[bsm][2 * ln] = c2;
        }
        __syncthreads();

        // ---- phase 3: s += c * u_hat (u_hat from registers, c as b128 rows) ----
        #pragma unroll
        for (int r = 0; r < 8; ++r) {
            float4 cc = *(const float4*)&ls[r + 8 * g][wave * NPW];
            sAcc[0][r] += cc.x * uh[0][r];
            sAcc[1][r] += cc.y * uh[1][r];
            sAcc[2][r] += cc.z * uh[2][r];
            sAcc[3][r] += cc.w * uh[3][r];
        }
        __syncthreads();

        wcur += WSTR_I;
    }

    // Reduce partial s across i-chunk workgroups (immediate-offset atomics).
    {
        float* ob = sbuf + ((size_t)(B0 + 8 * g) * Ncap + wave * NPW) * Dcap + ln;
        #pragma unroll
        for (int np = 0; np < NPW; ++np)
            #pragma unroll
            for (int r = 0; r < 8; ++r)
                unsafeAtomicAdd(&ob[(size_t)r * (Ncap * Dcap) + np * Dcap], sAcc[np][r]);
    }
}

// squash + Vacc accumulation (+ final sigmoid head on last pass)
__global__ void caps_update(float* __restrict__ Vacc, float* __restrict__ sbuf,
                            const float* __restrict__ dw, const float* __restrict__ db,
                            float* __restrict__ out, int write_out)
{
    int t = blockIdx.x * blockDim.x + threadIdx.x;
    if (t >= Bsz * Ncap * Dcap) return;
    float s = sbuf[t];
    float sq = s * s;
    float v = (sq / (1.0f + sq)) * rsqrtf(sq + EPSQ) * s;   // elementwise squash quirk
    if (write_out) {
        float z = v * dw[0] + db[0];
        out[t] = 1.0f / (1.0f + __expf(-z));
    } else {
        Vacc[t] += v;       // logits are linear in v: Vacc = sum of v's so far
        sbuf[t] = 0.0f;     // reset accumulator for next routing pass
    }
}

extern "C" void kernel_launch(void* const* d_in, const int* in_sizes, int n_in,
                              void* d_out, int out_size, void* d_ws, size_t ws_size,
                              hipStream_t stream)
{
    const float* x  = (const float*)d_in[0];   // [64,1152,8]
    const float* W  = (const float*)d_in[1];   // [32,1152,16,8]
    const float* dw = (const float*)d_in[2];   // [1,1]
    const float* db = (const float*)d_in[3];   // [1]
    float* out  = (float*)d_out;               // [64,32,16,1]
    float* Vacc = (float*)d_ws;                // 32768 floats
    float* sbuf = Vacc + Bsz * Ncap * Dcap;    // 32768 floats

    caps_zero<<<dim3(256), 256, 0, stream>>>(Vacc, 2 * Bsz * Ncap * Dcap);

    dim3 gridA(NCHUNK, Bsz / BT);   // (72, 4)
    for (int it = 0; it < 3; ++it) {
        caps_pass<<<gridA, 256, 0, stream>>>(x, W, Vacc, sbuf);
        caps_update<<<dim3(128), 256, 0, stream>>>(Vacc, sbuf, dw, db, out, it == 2 ? 1 : 0);
    }
}